// StackPointFeature_24893630447961
// MI455X (gfx1250) — compile-verified
//
#include <hip/hip_runtime.h>
#include <stdint.h>

typedef float v2f __attribute__((ext_vector_type(2)));
typedef float v8f __attribute__((ext_vector_type(8)));

#define NS1 16
#define NS2 32
#define R1  0.8f
#define R2  1.6f
#define CCH 64          // feature channels
#define OUTC (2*CCH)    // 128 output channels per query

__device__ __forceinline__ float bpermf(float v, int srcLane) {
    return __int_as_float(__builtin_amdgcn_ds_bpermute(srcLane << 2, __float_as_int(v)));
}

__global__ __launch_bounds__(32)
void StackPointFeature_ballgroup(const float* __restrict__ xyz,
                                 const int*   __restrict__ xyz_cnt,
                                 const float* __restrict__ new_xyz,
                                 const int*   __restrict__ new_cnt,
                                 const float* __restrict__ feats,
                                 float* __restrict__ out_xyz,
                                 float* __restrict__ out_feat,
                                 int B, int N, int M)
{
    const int lane = threadIdx.x;     // 0..31 (wave32)
    const int l16  = lane & 15;
    const int hi   = lane >> 4;       // 0: lanes 0-15, 1: lanes 16-31
    const int g0   = blockIdx.x * 16; // first query row of this tile

    // ---- batch lookup for this tile (uniform across wave) ----
    int cumM = 0, startN = 0, b = B - 1;
    for (int i = 0; i < B; ++i) {
        int mc = new_cnt[i];
        if (g0 < cumM + mc) { b = i; break; }
        cumM   += mc;
        startN += xyz_cnt[i];
    }
    const int cntN = xyz_cnt[b];

    // ---- load 16 query points (each half-wave loads the same 16 rows) ----
    const int  g        = g0 + l16;
    const bool rowValid = (g < M);
    const int  gq       = rowValid ? g : 0;
    float qx = new_xyz[gq*3+0];
    float qy = new_xyz[gq*3+1];
    float qz = new_xyz[gq*3+2];
    if (!rowValid) { qx = qy = qz = 1e18f; }   // never matches anything
    const float qn2 = qx*qx + qy*qy + qz*qz;

    // first tuple output: copy new_xyz through
    if (lane < 16 && rowValid) {
        out_xyz[g*3+0] = qx; out_xyz[g*3+1] = qy; out_xyz[g*3+2] = qz;
    }

    // ---- A matrix (16x4 f32): row m = (-2qx, -2qy, -2qz, 1) ----
    // ISA layout: lanes 0-15 hold K0(V0)/K1(V1); lanes 16-31 hold K2(V0)/K3(V1)
    v2f Amat;
    Amat.x = hi ? (-2.0f * qz) : (-2.0f * qx);
    Amat.y = hi ? 1.0f          : (-2.0f * qy);

    // ---- per-row thresholds for this lane's 8 D slots: row(r) = r + 8*hi ----
    float thr1[8], thr2[8];
#pragma unroll
    for (int r = 0; r < 8; ++r) {
        float qn2_row = bpermf(qn2, r + 8*hi);
        thr1[r] = R1*R1 - qn2_row;   // s < thr  <=>  d2 < r^2
        thr2[r] = R2*R2 - qn2_row;
    }

    // ---- accumulators: row m, lane owns channels {lane, lane+32} ----
    float2 acc1[16], acc2[16];
#pragma unroll
    for (int m = 0; m < 16; ++m) {
        acc1[m].x = acc1[m].y = -__builtin_inff();
        acc2[m].x = acc2[m].y = -__builtin_inff();
    }
    int c1 = 0, c2 = 0;   // neighbor counts; meaningful on lanes 0-15 (row = lane)

    const int nch = (cntN + 15) >> 4;

    // ---- software-pipelined point loads: prefetch chunk 0 ----
    float px, py, pz;
    {
        const bool pv = (l16 < cntN);
        const int  np = startN + (pv ? l16 : 0);
        px = xyz[np*3+0]; py = xyz[np*3+1]; pz = xyz[np*3+2];
        if (!pv) { px = py = pz = 3e18f; }
    }

    for (int chk = 0; chk < nch; ++chk) {
        // current chunk's coordinates
        const float cx = px, cy = py, cz = pz;

        // issue next chunk's loads early; s_wait lands next iteration
        {
            const int  nl2 = (chk+1)*16 + l16;
            const bool pv2 = (nl2 < cntN);
            const int  np2 = startN + (pv2 ? nl2 : 0);
            px = xyz[np2*3+0]; py = xyz[np2*3+1]; pz = xyz[np2*3+2];
            if (!pv2) { px = py = pz = 3e18f; }
        }

        const float pn2 = cx*cx + cy*cy + cz*cz;

        // B matrix (4x16 f32): col n = (px, py, pz, |p|^2), layout mirrors A
        v2f Bmat;
        Bmat.x = hi ? cz  : cx;
        Bmat.y = hi ? pn2 : cy;

        v8f Cz = {0.f,0.f,0.f,0.f,0.f,0.f,0.f,0.f};
        // s[m][n] = |p_n|^2 - 2 q_m . p_n
        v8f D = __builtin_amdgcn_wmma_f32_16x16x4_f32(
            false, Amat, false, Bmat, (short)0, Cz, false, false);

        // ballots: bit l of bal[r] => element (row r + 8*(l/16), col l%16) in radius
        uint32_t bal1[8], bal2[8];
#pragma unroll
        for (int r = 0; r < 8; ++r) {
            bal1[r] = __builtin_amdgcn_ballot_w32(D[r] < thr1[r]);
            bal2[r] = __builtin_amdgcn_ballot_w32(D[r] < thr2[r]);
        }

        // lane m<16 extracts its row's 16-bit column mask (no dynamic indexing)
        const int hsel = (lane >> 3) & 1;
        uint32_t m1 = 0, m2 = 0;
#pragma unroll
        for (int r = 0; r < 8; ++r) {
            uint32_t x1 = (bal1[r] >> (hsel*16)) & 0xFFFFu;
            uint32_t x2 = (bal2[r] >> (hsel*16)) & 0xFFFFu;
            bool sel = ((lane & 7) == r);
            m1 = sel ? x1 : m1;
            m2 = sel ? x2 : m2;
        }
        if (lane >= 16) { m1 = 0; m2 = 0; }

        // take the first (lowest-index) remaining slots for each radius.
        // Common cases are branch-free: quota untouched (take all) or full (take none).
        uint32_t tm1, tm2;
        {
            int avail = __builtin_popcount(m1);
            int need  = NS1 - c1;
            int take  = avail < need ? avail : need;
            tm1 = m1;
            if (avail > need) {            // rare: quota-crossing chunk
                tm1 = 0; uint32_t mm = m1;
                for (int i = 0; i < take; ++i) { tm1 |= mm & (0u - mm); mm &= mm - 1; }
            }
            c1 += take;
        }
        {
            int avail = __builtin_popcount(m2);
            int need  = NS2 - c2;
            int take  = avail < need ? avail : need;
            tm2 = m2;
            if (avail > need) {
                tm2 = 0; uint32_t mm = m2;
                for (int i = 0; i < take; ++i) { tm2 |= mm & (0u - mm); mm &= mm - 1; }
            }
            c2 += take;
        }

        // wave-cooperative feature max-accumulate (uniform loops);
        // skip entirely when no row took anything this chunk
        if (__builtin_amdgcn_ballot_w32((tm1 | tm2) != 0) != 0u) {
            const float* fbase = feats + (size_t)(startN + chk*16) * CCH;
#pragma unroll
            for (int m = 0; m < 16; ++m) {
                uint32_t t1 = (uint32_t)__builtin_amdgcn_readlane((int)tm1, m);
                uint32_t t2 = (uint32_t)__builtin_amdgcn_readlane((int)tm2, m);
                uint32_t u  = t1 | t2;
                while (u) {
                    int nl = __builtin_ctz(u); u &= u - 1;
                    const float* fp = fbase + (size_t)nl * CCH;
                    float fa = fp[lane];
                    float fb = fp[lane + 32];
                    uint32_t bit = 1u << nl;
                    if (t1 & bit) { acc1[m].x = fmaxf(acc1[m].x, fa); acc1[m].y = fmaxf(acc1[m].y, fb); }
                    if (t2 & bit) { acc2[m].x = fmaxf(acc2[m].x, fa); acc2[m].y = fmaxf(acc2[m].y, fb); }
                }
            }
        }

        // uniform early exit once every row is full for both radii
        int mydone = (lane < 16) ? ((c1 >= NS1) && (c2 >= NS2)) : 1;
        if (__builtin_amdgcn_ballot_w32(mydone != 0) == 0xFFFFFFFFu) break;
    }

    // ---- write pooled features: row gg -> [acc1 ch0..63 | acc2 ch0..63] ----
#pragma unroll
    for (int m = 0; m < 16; ++m) {
        int gg = g0 + m;
        if (gg >= M) continue;  // uniform
        int cc1 = __builtin_amdgcn_readlane(c1, m);
        int cc2 = __builtin_amdgcn_readlane(c2, m);
        float o0 = (cc1 > 0) ? acc1[m].x : 0.0f;
        float o1 = (cc1 > 0) ? acc1[m].y : 0.0f;
        float o2 = (cc2 > 0) ? acc2[m].x : 0.0f;
        float o3 = (cc2 > 0) ? acc2[m].y : 0.0f;
        float* op = out_feat + (size_t)gg * OUTC;
        op[lane]      = o0;
        op[lane + 32] = o1;
        op[lane + 64] = o2;
        op[lane + 96] = o3;
    }
}

extern "C" void kernel_launch(void* const* d_in, const int* in_sizes, int n_in,
                              void* d_out, int out_size, void* d_ws, size_t ws_size,
                              hipStream_t stream) {
    (void)n_in; (void)d_ws; (void)ws_size; (void)out_size;
    const float* xyz      = (const float*)d_in[0];
    const int*   xyz_cnt  = (const int*)  d_in[1];
    const float* new_xyz  = (const float*)d_in[2];
    const int*   new_cnt  = (const int*)  d_in[3];
    const float* feats    = (const float*)d_in[4];

    const int N = in_sizes[0] / 3;
    const int B = in_sizes[1];
    const int M = in_sizes[2] / 3;

    float* out      = (float*)d_out;
    float* out_xyz  = out;                    // tuple part 0: new_xyz (M*3)
    float* out_feat = out + (size_t)M * 3;    // tuple part 1: features (M*128)

    const int tiles = (M + 15) / 16;
    hipLaunchKernelGGL(StackPointFeature_ballgroup, dim3(tiles), dim3(32), 0, stream,
                       xyz, xyz_cnt, new_xyz, new_cnt, feats,
                       out_xyz, out_feat, B, N, M);
}